// DecoderRNNsearch_28862180229238
// MI455X (gfx1250) — compile-verified
//
#include <hip/hip_runtime.h>
#include <hip/hip_bf16.h>
#include <stdint.h>

// Problem dims (compile-time, match reference)
static constexpr int Bsz  = 64;     // batch
static constexpr int Ssz  = 64;     // tgt seq len
static constexpr int Lsz  = 64;     // ctx len
static constexpr int Hsz  = 1024;   // hidden
static constexpr int Csz  = 2048;   // ctx dim
static constexpr int Esz  = 512;    // emb dim
static constexpr int OUTW = Hsz + Esz + Csz;   // 3584 output row width

typedef __attribute__((ext_vector_type(2))) float v2f;
typedef __attribute__((ext_vector_type(8))) float v8f;

// ---------------------------------------------------------------------------
// Generic fp32 WMMA GEMM:  Cout[M,N] = A1[M,K1] * B1[N,K1]^T
//                                    (+ A2[M,K2] * B2[N,K2]^T)
//                                    (+ bias[N]) (+ Cin[M,N])
// Weights B are row-major [N, K] (PyTorch layout), arbitrary leading dims.
// One wave32 computes a 64x16 output tile: 4 accumulators share each B
// fragment (4x less weight traffic, 4 independent WMMA chains to hide XDL
// latency). V_WMMA_F32_16X16X4_F32 over K in steps of 4.
// blockDim.x == 32 (EXEC all ones for WMMA). grid = (N/16, M/64).
// ---------------------------------------------------------------------------
__global__ void gemm_wmma_f32(const float* __restrict__ A1, long lda1,
                              const float* __restrict__ B1, long ldb1, int K1,
                              const float* __restrict__ A2, long lda2,
                              const float* __restrict__ B2, long ldb2, int K2,
                              const float* __restrict__ bias,
                              const float* __restrict__ Cin, long ldcin,
                              float* __restrict__ Cout, long ldc)
{
    const int lane = threadIdx.x;          // 0..31
    const int n0   = blockIdx.x * 16;
    const int m0   = blockIdx.y * 64;
    const int r15  = lane & 15;            // A row (M) and B row (N) within tile
    const int kh   = (lane >> 4) << 1;     // K half-select: 0 or 2

    v8f zero = {};
    v8f acc[4] = {zero, zero, zero, zero};

    {
        const float* arow = A1 + (size_t)(m0 + r15) * lda1 + kh;
        const float* brow = B1 + (size_t)(n0 + r15) * ldb1 + kh;
        for (int k = 0; k < K1; k += 4) {
            v2f b = *(const v2f*)(brow + k);
            #pragma unroll
            for (int t = 0; t < 4; ++t) {
                v2f a = *(const v2f*)(arow + (size_t)(t * 16) * lda1 + k);
                acc[t] = __builtin_amdgcn_wmma_f32_16x16x4_f32(
                             false, a, false, b, (short)0, acc[t], false, false);
            }
        }
    }
    if (A2) {
        const float* arow = A2 + (size_t)(m0 + r15) * lda2 + kh;
        const float* brow = B2 + (size_t)(n0 + r15) * ldb2 + kh;
        for (int k = 0; k < K2; k += 4) {
            v2f b = *(const v2f*)(brow + k);
            #pragma unroll
            for (int t = 0; t < 4; ++t) {
                v2f a = *(const v2f*)(arow + (size_t)(t * 16) * lda2 + k);
                acc[t] = __builtin_amdgcn_wmma_f32_16x16x4_f32(
                             false, a, false, b, (short)0, acc[t], false, false);
            }
        }
    }

    // C/D layout: VGPR i -> row i + 8*(lane>>4), col lane&15
    const int cn = n0 + r15;
    const int rb = (lane >> 4) * 8;
    const float bv = bias ? bias[cn] : 0.0f;
    #pragma unroll
    for (int t = 0; t < 4; ++t) {
        #pragma unroll
        for (int i = 0; i < 8; ++i) {
            const int cm = m0 + t * 16 + rb + i;
            float val = acc[t][i] + bv;
            if (Cin) val += Cin[(size_t)cm * ldcin + cn];
            Cout[(size_t)cm * ldc + cn] = val;
        }
    }
}

// ---------------------------------------------------------------------------
// Gather embeddings for all (b,s) directly into the output buffer columns
// [H, H+E). Output row index for (b,s) is b*S+s == flat index of tgt.
// ---------------------------------------------------------------------------
__global__ void gather_embed_kernel(const int* __restrict__ tgt,
                                    const float* __restrict__ emb,
                                    float* __restrict__ out)
{
    const int row = blockIdx.x;                 // b*S + s
    const int t   = tgt[row];
    const float* src = emb + (size_t)t * Esz;
    float* dst = out + (size_t)row * OUTW + Hsz;
    for (int j = threadIdx.x; j < Esz; j += blockDim.x) dst[j] = src[j];
}

// ---------------------------------------------------------------------------
// Per-step attention: score[b,l] = sum_h v[h]*tanh(ctx_proj[b,l,h]+q[b,h]);
// softmax over l; ctx_t[b,c] = sum_l attn[l]*ctx[b,l,c].
// Also writes out[b*S+s, 0:H] = h (pre-update) and out[.., H+E:] = ctx_t.
// One 256-thread block per batch element.
// ---------------------------------------------------------------------------
__global__ void attn_kernel(const float* __restrict__ ctx_proj,
                            const float* __restrict__ q,
                            const float* __restrict__ vvec,
                            const float* __restrict__ ctx,
                            const float* __restrict__ h,
                            float* __restrict__ ctxt,
                            float* __restrict__ out, int s)
{
    const int b    = blockIdx.x;
    const int tid  = threadIdx.x;        // 256 threads = 8 waves
    const int wave = tid >> 5;
    const int lane = tid & 31;
    __shared__ float sc[Lsz];

    // scores: each wave handles l = wave, wave+8, ...
    for (int l = wave; l < Lsz; l += 8) {
        const float* cp = ctx_proj + ((size_t)b * Lsz + l) * Hsz;
        const float* qb = q + (size_t)b * Hsz;
        float sum = 0.0f;
        for (int hh = lane; hh < Hsz; hh += 32)
            sum += vvec[hh] * tanhf(cp[hh] + qb[hh]);
        for (int off = 16; off > 0; off >>= 1)
            sum += __shfl_xor(sum, off, 32);
        if (lane == 0) sc[l] = sum;
    }
    __syncthreads();

    // softmax over 64 values (tiny; serial in thread 0)
    if (tid == 0) {
        float mx = sc[0];
        for (int l = 1; l < Lsz; ++l) mx = fmaxf(mx, sc[l]);
        float ssum = 0.0f;
        for (int l = 0; l < Lsz; ++l) { float e = expf(sc[l] - mx); sc[l] = e; ssum += e; }
        float inv = 1.0f / ssum;
        for (int l = 0; l < Lsz; ++l) sc[l] *= inv;
    }
    __syncthreads();

    const size_t orow = ((size_t)b * Ssz + s) * OUTW;

    // ctx_t and output ctx columns
    for (int c = tid; c < Csz; c += 256) {
        const float* cb = ctx + (size_t)b * Lsz * Csz + c;
        float acc = 0.0f;
        for (int l = 0; l < Lsz; ++l) acc += sc[l] * cb[(size_t)l * Csz];
        ctxt[(size_t)b * Csz + c] = acc;
        out[orow + Hsz + Esz + c] = acc;
    }
    // pre-update hidden state into output columns [0,H)
    for (int j = tid; j < Hsz; j += 256)
        out[orow + j] = h[(size_t)b * Hsz + j];
}

// ---------------------------------------------------------------------------
// GRU pointwise update (PyTorch semantics), in-place on h.
// gi/gh are [B, 3H] with gate order r, z, n.
// ---------------------------------------------------------------------------
__global__ void gru_kernel(const float* __restrict__ gi,
                           const float* __restrict__ gh,
                           float* __restrict__ h)
{
    const int b = blockIdx.x;
    const size_t g0 = (size_t)b * 3 * Hsz;
    for (int j = threadIdx.x; j < Hsz; j += blockDim.x) {
        const float ir = gi[g0 + j],            hr = gh[g0 + j];
        const float iz = gi[g0 + Hsz + j],      hz = gh[g0 + Hsz + j];
        const float in_ = gi[g0 + 2 * Hsz + j], hn = gh[g0 + 2 * Hsz + j];
        const float r = 1.0f / (1.0f + expf(-(ir + hr)));
        const float z = 1.0f / (1.0f + expf(-(iz + hz)));
        const float n = tanhf(in_ + r * hn);
        const float hp = h[(size_t)b * Hsz + j];
        h[(size_t)b * Hsz + j] = (1.0f - z) * n + z * hp;
    }
}

__global__ void copy_kernel(const float* __restrict__ src,
                            float* __restrict__ dst, int n)
{
    const int i = blockIdx.x * blockDim.x + threadIdx.x;
    if (i < n) dst[i] = src[i];
}

// ---------------------------------------------------------------------------
extern "C" void kernel_launch(void* const* d_in, const int* in_sizes, int n_in,
                              void* d_out, int out_size, void* d_ws, size_t ws_size,
                              hipStream_t stream)
{
    (void)in_sizes; (void)n_in; (void)out_size; (void)ws_size;

    const int*   tgt  = (const int*)  d_in[0];
    const float* ctx  = (const float*)d_in[1];
    const float* h0   = (const float*)d_in[2];
    const float* emb  = (const float*)d_in[3];
    const float* Wc   = (const float*)d_in[4];
    const float* Wq   = (const float*)d_in[5];
    const float* bq   = (const float*)d_in[6];
    const float* vvec = (const float*)d_in[7];
    const float* W_ih = (const float*)d_in[8];
    const float* W_hh = (const float*)d_in[9];
    const float* b_ih = (const float*)d_in[10];
    const float* b_hh = (const float*)d_in[11];
    float* out = (float*)d_out;

    // Workspace carve-up (floats); ~19.4 MB total
    float* ws = (float*)d_ws;
    float* ctx_proj = ws;  ws += (size_t)Bsz * Lsz * Hsz;   // [B,L,H]
    float* hbuf     = ws;  ws += (size_t)Bsz * Hsz;         // [B,H]
    float* qbuf     = ws;  ws += (size_t)Bsz * Hsz;         // [B,H]
    float* ctxt     = ws;  ws += (size_t)Bsz * Csz;         // [B,C]
    float* gib      = ws;  ws += (size_t)Bsz * 3 * Hsz;     // [B,3H]
    float* ghb      = ws;  ws += (size_t)Bsz * 3 * Hsz;     // [B,3H]

    // h = hidden0 (do not mutate inputs)
    hipMemcpyAsync(hbuf, h0, (size_t)Bsz * Hsz * sizeof(float),
                   hipMemcpyDeviceToDevice, stream);

    // Embedding gather straight into output columns [H, H+E) for all rows
    gather_embed_kernel<<<Bsz * Ssz, 256, 0, stream>>>(tgt, emb, out);

    // ctx_proj = ctx @ Wc^T : M=B*L=4096, N=H=1024, K=C=2048
    gemm_wmma_f32<<<dim3(Hsz / 16, (Bsz * Lsz) / 64), 32, 0, stream>>>(
        ctx, Csz, Wc, Csz, Csz,
        nullptr, 0, nullptr, 0, 0,
        nullptr, nullptr, 0, ctx_proj, Hsz);

    for (int s = 0; s < Ssz; ++s) {
        // q = h @ Wq^T + bq : M=64, N=1024, K=1024
        gemm_wmma_f32<<<dim3(Hsz / 16, Bsz / 64), 32, 0, stream>>>(
            hbuf, Hsz, Wq, Hsz, Hsz,
            nullptr, 0, nullptr, 0, 0,
            bq, nullptr, 0, qbuf, Hsz);

        // attention + write pre-update h and ctx_t into output rows
        attn_kernel<<<Bsz, 256, 0, stream>>>(ctx_proj, qbuf, vvec, ctx,
                                             hbuf, ctxt, out, s);

        // gi = emb_t @ W_ih[:, :E]^T + ctx_t @ W_ih[:, E:]^T + b_ih
        //   emb_t read from output buffer rows (cols [H,H+E)), lda = S*OUTW
        gemm_wmma_f32<<<dim3(3 * Hsz / 16, Bsz / 64), 32, 0, stream>>>(
            out + (size_t)s * OUTW + Hsz, (long)Ssz * OUTW,
            W_ih, Esz + Csz, Esz,
            ctxt, Csz, W_ih + Esz, Esz + Csz, Csz,
            b_ih, nullptr, 0, gib, 3 * Hsz);

        // gh = h @ W_hh^T + b_hh : M=64, N=3072, K=1024
        gemm_wmma_f32<<<dim3(3 * Hsz / 16, Bsz / 64), 32, 0, stream>>>(
            hbuf, Hsz, W_hh, Hsz, Hsz,
            nullptr, 0, nullptr, 0, 0,
            b_hh, nullptr, 0, ghb, 3 * Hsz);

        // h <- GRU(h, gi, gh)  (in place)
        gru_kernel<<<Bsz, 256, 0, stream>>>(gib, ghb, hbuf);
    }

    // h_last appended after outputs
    copy_kernel<<<(Bsz * Hsz + 255) / 256, 256, 0, stream>>>(
        hbuf, out + (size_t)Bsz * Ssz * OUTW, Bsz * Hsz);
}